// GConvoLayer_19078244729205
// MI455X (gfx1250) — compile-verified
//
#include <hip/hip_runtime.h>

#define NN 100000
#define NE 3200000
#define D 128
#define NSTRIPES (NN / 16)        // 6250
#define WPB 8                     // waves per block
#define HPAD 132                  // padded LDS row pitch for h (conflict-free)

typedef float v2f __attribute__((ext_vector_type(2)));
typedef float v8f __attribute__((ext_vector_type(8)));

// ---------------- Kernel 1: conv = X (vectorized copy) ----------------
__global__ __launch_bounds__(256) void k_copy_x(const float4* __restrict__ X,
                                                float4* __restrict__ conv, int n4) {
  int i = blockIdx.x * 256 + threadIdx.x;
  if (i < n4) conv[i] = X[i];
}

// ---------------- Kernel 2: conv[dst] += X[src] (edge scatter) ----------------
// 32 threads per edge, float4 per thread -> 4 f32 atomics each.
__global__ __launch_bounds__(256) void k_edge_scatter(const float* __restrict__ X,
                                                      const int* __restrict__ dst,
                                                      const int* __restrict__ src,
                                                      float* conv) {
  long long idx = (long long)blockIdx.x * 256 + threadIdx.x;
  int e = (int)(idx >> 5);
  int c = ((int)idx & 31) * 4;
  if (e >= NE) return;
  int s = src[e];
  int d = dst[e];
  float4 v = *(const float4*)(X + (size_t)s * D + c);
  float* p = conv + (size_t)d * D + c;
  unsafeAtomicAdd(p + 0, v.x);
  unsafeAtomicAdd(p + 1, v.y);
  unsafeAtomicAdd(p + 2, v.z);
  unsafeAtomicAdd(p + 3, v.w);
}

// ---------------- Kernel 3: fused out = relu(relu(conv@W1+b1)@W2+b2) ----------------
// One wave per 16-row stripe. Weights staged in LDS in B-fragment layout.
// h bounced through LDS (D-frag -> A-frag relayout), never touches global.
__global__ __launch_bounds__(256) void k_gin_mlp(const float* conv,
                                                 const float* __restrict__ w1,
                                                 const float* __restrict__ b1,
                                                 const float* __restrict__ w2,
                                                 const float* __restrict__ b2,
                                                 float* out) {
  __shared__ float lw1[16384];            // 64 KB: W1 B-fragments
  __shared__ float lw2[16384];            // 64 KB: W2 B-fragments
  __shared__ float lh[WPB * 16 * HPAD];   // 66 KB: per-wave h bounce

  const int tid = threadIdx.x;

  // Stage W1/W2 into B-fragment layout:
  // pair index i = (kt*8 + nt)*32 + ln ; lane ln holds W[kt*4 + (ln>>4)*2 + {0,1}][nt*16 + (ln&15)]
  for (int i = tid; i < 8192; i += 256) {
    const int ln = i & 31;
    const int tile = i >> 5;              // kt*8 + nt
    const int kt = tile >> 3, nt = tile & 7;
    const int kr = kt * 4 + ((ln >> 4) << 1);
    const int col = nt * 16 + (ln & 15);
    lw1[2 * i + 0] = w1[(kr + 0) * D + col];
    lw1[2 * i + 1] = w1[(kr + 1) * D + col];
    lw2[2 * i + 0] = w2[(kr + 0) * D + col];
    lw2[2 * i + 1] = w2[(kr + 1) * D + col];
  }
  __syncthreads();

  const int wave = tid >> 5;
  const int ln = tid & 31;
  const int stripe = blockIdx.x * WPB + wave;
  if (stripe >= NSTRIPES) return;         // whole wave exits together: EXEC stays all-1s
  const int row0 = stripe * 16;
  const int lcol = ln & 15;
  const int kb = (ln >> 4) << 1;          // K sub-offset: 0 or 2
  const int hrow0 = (ln >> 4) << 3;       // D-fragment row base: 0 or 8

  // ---- GEMM1: h = relu(conv @ W1 + b1) ----
  v8f acc[8] = {};
  const float* aptr = conv + (size_t)(row0 + lcol) * D + kb;
#pragma unroll 4
  for (int kt = 0; kt < 32; ++kt) {
    v2f a = *(const v2f*)(aptr + kt * 4);
#pragma unroll
    for (int nt = 0; nt < 8; ++nt) {
      v2f b = *(const v2f*)(&lw1[((kt * 8 + nt) * 32 + ln) * 2]);
      acc[nt] = __builtin_amdgcn_wmma_f32_16x16x4_f32(
          false, a, false, b, (short)0, acc[nt], false, false);
    }
  }

  // bias + relu, D-fragment -> row-major LDS stripe (padded pitch)
  float* hb = &lh[wave * 16 * HPAD];
#pragma unroll
  for (int nt = 0; nt < 8; ++nt) {
    const int col = nt * 16 + lcol;
    const float bias = b1[col];
#pragma unroll
    for (int r = 0; r < 8; ++r) {
      float h = acc[nt][r] + bias;
      hb[(hrow0 + r) * HPAD + col] = h > 0.0f ? h : 0.0f;
    }
  }
  // same-wave LDS RAW across lanes: LDS is in-order per wave; just drain DScnt
  asm volatile("s_wait_dscnt 0x0" ::: "memory");

  // ---- GEMM2: out = relu(h @ W2 + b2) ----
  v8f acc2[8] = {};
  const float* haptr = hb + lcol * HPAD + kb;
#pragma unroll 4
  for (int kt = 0; kt < 32; ++kt) {
    v2f a = *(const v2f*)(haptr + kt * 4);
#pragma unroll
    for (int nt = 0; nt < 8; ++nt) {
      v2f b = *(const v2f*)(&lw2[((kt * 8 + nt) * 32 + ln) * 2]);
      acc2[nt] = __builtin_amdgcn_wmma_f32_16x16x4_f32(
          false, a, false, b, (short)0, acc2[nt], false, false);
    }
  }

#pragma unroll
  for (int nt = 0; nt < 8; ++nt) {
    const int col = nt * 16 + lcol;
    const float bias = b2[col];
#pragma unroll
    for (int r = 0; r < 8; ++r) {
      float o = acc2[nt][r] + bias;
      out[(size_t)(row0 + hrow0 + r) * D + col] = o > 0.0f ? o : 0.0f;
    }
  }
}

// ---------------- Launch ----------------
extern "C" void kernel_launch(void* const* d_in, const int* in_sizes, int n_in,
                              void* d_out, int out_size, void* d_ws, size_t ws_size,
                              hipStream_t stream) {
  const float* X  = (const float*)d_in[0];
  const int* refA = (const int*)d_in[1];   // dst per edge
  const int* refB = (const int*)d_in[2];   // src per edge
  const float* w1 = (const float*)d_in[3];
  const float* b1 = (const float*)d_in[4];
  const float* w2 = (const float*)d_in[5];
  const float* b2 = (const float*)d_in[6];
  float* out = (float*)d_out;

  const size_t convBytes = (size_t)NN * D * sizeof(float);
  // conv scratch: prefer d_ws; d_out is a safe fallback (each wave in k_gin_mlp
  // reads its whole conv stripe before writing the same stripe of out).
  float* conv = (ws_size >= convBytes) ? (float*)d_ws : out;

  const int n4 = NN * D / 4;  // 3,200,000 float4s
  k_copy_x<<<(n4 + 255) / 256, 256, 0, stream>>>((const float4*)X, (float4*)conv, n4);

  const long long ethreads = (long long)NE * 32;  // 102,400,000
  k_edge_scatter<<<(int)((ethreads + 255) / 256), 256, 0, stream>>>(X, refA, refB, conv);

  k_gin_mlp<<<(NSTRIPES + WPB - 1) / WPB, 256, 0, stream>>>(conv, w1, b1, w2, b2, out);
}